// GIN_concat_22625887715638
// MI455X (gfx1250) — compile-verified
//
#include <hip/hip_runtime.h>

typedef float v2f __attribute__((ext_vector_type(2)));
typedef float v8f __attribute__((ext_vector_type(8)));
typedef float f4  __attribute__((ext_vector_type(4)));

#define N_NODES 50000
#define N_EDGES 800000
#define FDIM    128
#define EDIM    100
#define K1      (FDIM + EDIM)   /* 228 */
#define K1P     232             /* padded to multiple of 8 */
#define GNUM    512
#define NLAYERS 2
#define BN_EPS  1e-5f

__device__ __forceinline__ v8f wmma4(v2f a, v2f b, v8f c) {
  // V_WMMA_F32_16X16X4_F32: D = A(16x4,f32) * B(4x16,f32) + C(16x16,f32)
  return __builtin_amdgcn_wmma_f32_16x16x4_f32(false, a, false, b, (short)0, c,
                                               false, false);
}

// ---------------------------------------------------------------------------
// Repack W[K][128] into WMMA fragment order, zero-padded to Kp rows:
//   Wp[(kb*8 + ntile)*32 + lane] = f4{ W[kb*8+bk][n], W[kb*8+bk+1][n],
//                                      W[kb*8+4+bk][n], W[kb*8+5+bk][n] }
//   with ln=lane&15, hi=lane>>4, bk=2*hi, n=ntile*16+ln.
// One b128 per lane then feeds two K-steps of a wave's WMMA loop, fully
// coalesced (consecutive lanes -> consecutive 16B).
// ---------------------------------------------------------------------------
__global__ __launch_bounds__(256)
void repack_w_kernel(const float* __restrict__ W, float* __restrict__ Wp,
                     int K, int Kp)
{
  int idx = blockIdx.x * 256 + threadIdx.x;     // (kb, ntile, lane)
  int total = Kp * 32;                          // (Kp/8)*8*32
  if (idx >= total) return;
  int lane = idx & 31;
  int nt   = (idx >> 5) & 7;
  int kb   = idx >> 8;
  int ln = lane & 15, hi = lane >> 4, bk = hi * 2;
  int n  = nt * 16 + ln;
  int k0 = kb * 8 + bk;
  int k2 = kb * 8 + 4 + bk;
  f4 v;
  v.x = (k0     < K) ? W[(size_t)(k0)     * FDIM + n] : 0.f;
  v.y = (k0 + 1 < K) ? W[(size_t)(k0 + 1) * FDIM + n] : 0.f;
  v.z = (k2     < K) ? W[(size_t)(k2)     * FDIM + n] : 0.f;
  v.w = (k2 + 1 < K) ? W[(size_t)(k2 + 1) * FDIM + n] : 0.f;
  ((f4*)Wp)[idx] = v;
}

// ---------------------------------------------------------------------------
// Edge MLP: m = ReLU(concat(x[src],eattr) @ W1 + b1) @ W2 + b2
//           aggr[dst] += m  (skip self loops)
// Block: 32 edges (2 M-tiles), 256 threads = 8 waves; wave owns one 16-col
// N tile for both M-tiles (each B fragment feeds 2 WMMAs).
// ---------------------------------------------------------------------------
__global__ __launch_bounds__(256)
void edge_mlp_kernel(const float* __restrict__ x, const float* __restrict__ eattr,
                     const int* __restrict__ src, const int* __restrict__ dst,
                     const float* __restrict__ W1p, const float* __restrict__ b1,
                     const float* __restrict__ W2p, const float* __restrict__ b2,
                     float* __restrict__ aggr)
{
  const int ZS = 236;   // padded z row stride (floats), 16B aligned rows
  const int HS = 132;   // padded hidden row stride
  __shared__ float z[32 * ZS];
  __shared__ float hid[32 * HS];
  __shared__ int s_src[32], s_dst[32];

  const int tid = threadIdx.x;
  const int e0  = blockIdx.x * 32;

  if (tid < 32) {
    int e = e0 + tid;
    s_src[tid] = (e < N_EDGES) ? src[e] : 0;
    s_dst[tid] = (e < N_EDGES) ? dst[e] : 0;
  }
  __syncthreads();

  // stage x[src[e]] rows: 32 rows x 32 float4
  for (int i = tid; i < 32 * 32; i += 256) {
    int r = i >> 5, c4 = i & 31;
    const f4* p = (const f4*)(x + (size_t)s_src[r] * FDIM);
    *(f4*)(&z[r * ZS + c4 * 4]) = p[c4];
  }
  // stage edge_attr rows: 32 rows x 25 float4 (100 floats, 16B aligned/row)
  for (int i = tid; i < 32 * 25; i += 256) {
    int r = i / 25, c4 = i % 25;
    int e = e0 + r;
    if (e < N_EDGES) {
      const f4* p = (const f4*)(eattr + (size_t)e * EDIM);
      *(f4*)(&z[r * ZS + FDIM + c4 * 4]) = p[c4];
    }
  }
  // zero K padding columns 228..231 (match zero rows in repacked W1)
  if (tid < 32) *(f4*)(&z[tid * ZS + K1]) = f4{0.f, 0.f, 0.f, 0.f};
  __syncthreads();

  const int lane = tid & 31;
  const int wv   = tid >> 5;
  const int n0   = wv * 16;
  const int ln   = lane & 15;      // A row (M) / B col (N) index
  const int hi   = lane >> 4;
  const int bk   = hi * 2;         // K sub-offset within the x4 step

  // ---- GEMM1: hidden = ReLU(z @ W1 + b1), K = 232 (padded) ----
  v8f acc0 = {0.f,0.f,0.f,0.f,0.f,0.f,0.f,0.f};
  v8f acc1 = {0.f,0.f,0.f,0.f,0.f,0.f,0.f,0.f};
  {
    const f4* W1p4 = (const f4*)W1p;
    #pragma unroll 4
    for (int kb = 0; kb < K1P / 8; ++kb) {
      f4 bq = W1p4[(size_t)(kb * 8 + wv) * 32 + lane];
      v2f b01 = {bq.x, bq.y};
      v2f b23 = {bq.z, bq.w};
      int kk = kb * 8;
      v2f a0 = *(const v2f*)(&z[ln * ZS + kk + bk]);
      v2f a1 = *(const v2f*)(&z[(16 + ln) * ZS + kk + bk]);
      acc0 = wmma4(a0, b01, acc0);
      acc1 = wmma4(a1, b01, acc1);
      a0 = *(const v2f*)(&z[ln * ZS + kk + 4 + bk]);
      a1 = *(const v2f*)(&z[(16 + ln) * ZS + kk + 4 + bk]);
      acc0 = wmma4(a0, b23, acc0);
      acc1 = wmma4(a1, b23, acc1);
    }
  }
  {
    float bb = b1[n0 + ln];
    #pragma unroll
    for (int v = 0; v < 8; ++v) {
      hid[(v + 8 * hi) * HS + n0 + ln]      = fmaxf(acc0[v] + bb, 0.f);
      hid[(16 + v + 8 * hi) * HS + n0 + ln] = fmaxf(acc1[v] + bb, 0.f);
    }
  }
  __syncthreads();

  // ---- GEMM2: m = hidden @ W2 + b2, K = 128 ----
  acc0 = v8f{0.f,0.f,0.f,0.f,0.f,0.f,0.f,0.f};
  acc1 = v8f{0.f,0.f,0.f,0.f,0.f,0.f,0.f,0.f};
  {
    const f4* W2p4 = (const f4*)W2p;
    #pragma unroll 4
    for (int kb = 0; kb < FDIM / 8; ++kb) {
      f4 bq = W2p4[(size_t)(kb * 8 + wv) * 32 + lane];
      v2f b01 = {bq.x, bq.y};
      v2f b23 = {bq.z, bq.w};
      int kk = kb * 8;
      v2f a0 = *(const v2f*)(&hid[ln * HS + kk + bk]);
      v2f a1 = *(const v2f*)(&hid[(16 + ln) * HS + kk + bk]);
      acc0 = wmma4(a0, b01, acc0);
      acc1 = wmma4(a1, b01, acc1);
      a0 = *(const v2f*)(&hid[ln * HS + kk + 4 + bk]);
      a1 = *(const v2f*)(&hid[(16 + ln) * HS + kk + 4 + bk]);
      acc0 = wmma4(a0, b23, acc0);
      acc1 = wmma4(a1, b23, acc1);
    }
  }
  {
    float bb = b2[n0 + ln];
    #pragma unroll
    for (int v = 0; v < 8; ++v) {
      int r = v + 8 * hi;
      int e = e0 + r;
      if (e < N_EDGES && s_src[r] != s_dst[r])
        atomicAdd(&aggr[(size_t)s_dst[r] * FDIM + n0 + ln], acc0[v] + bb);
      r = 16 + v + 8 * hi;
      e = e0 + r;
      if (e < N_EDGES && s_src[r] != s_dst[r])
        atomicAdd(&aggr[(size_t)s_dst[r] * FDIM + n0 + ln], acc1[v] + bb);
    }
  }
}

// ---------------------------------------------------------------------------
// Generic 32-row GEMM:  Y = post( pre(A [+Aadd]) @ W + bias ),  K = N = 128
//  W given in repacked fragment order (Kp=128).
//  pre  : optional per-column affine + ReLU (fused BN apply)
//  post : optional ReLU
//  psum/psq: optional per-block BN partial sums of Y columns
// ---------------------------------------------------------------------------
__global__ __launch_bounds__(256)
void gemm32_kernel(const float* __restrict__ A, const float* __restrict__ Aadd,
                   const float* __restrict__ preScale, const float* __restrict__ preShift,
                   const float* __restrict__ Wp, const float* __restrict__ bias,
                   float* __restrict__ Y,
                   float* __restrict__ psum, float* __restrict__ psq,
                   int postRelu, int M)
{
  const int AS = 132;
  __shared__ float a_lds[32 * AS];
  const int tid = threadIdx.x;
  const int r0  = blockIdx.x * 32;

  for (int i = tid; i < 32 * 32; i += 256) {
    int r = i >> 5, c4 = i & 31;
    int gr = r0 + r;
    f4 v = {0.f, 0.f, 0.f, 0.f};
    if (gr < M) {
      v = *(const f4*)(A + (size_t)gr * FDIM + c4 * 4);
      if (Aadd) {
        f4 w = *(const f4*)(Aadd + (size_t)gr * FDIM + c4 * 4);
        v.x += w.x; v.y += w.y; v.z += w.z; v.w += w.w;
      }
      if (preScale) {
        int c = c4 * 4;
        v.x = fmaxf(v.x * preScale[c + 0] + preShift[c + 0], 0.f);
        v.y = fmaxf(v.y * preScale[c + 1] + preShift[c + 1], 0.f);
        v.z = fmaxf(v.z * preScale[c + 2] + preShift[c + 2], 0.f);
        v.w = fmaxf(v.w * preScale[c + 3] + preShift[c + 3], 0.f);
      }
    }
    *(f4*)(&a_lds[r * AS + c4 * 4]) = v;
  }
  __syncthreads();

  const int lane = tid & 31;
  const int wv   = tid >> 5;
  const int n0   = wv * 16;
  const int ln   = lane & 15;
  const int hi   = lane >> 4;
  const int bk   = hi * 2;

  v8f acc0 = {0.f,0.f,0.f,0.f,0.f,0.f,0.f,0.f};
  v8f acc1 = {0.f,0.f,0.f,0.f,0.f,0.f,0.f,0.f};
  {
    const f4* Wp4 = (const f4*)Wp;
    #pragma unroll 4
    for (int kb = 0; kb < FDIM / 8; ++kb) {
      f4 bq = Wp4[(size_t)(kb * 8 + wv) * 32 + lane];
      v2f b01 = {bq.x, bq.y};
      v2f b23 = {bq.z, bq.w};
      int kk = kb * 8;
      v2f a0 = *(const v2f*)(&a_lds[ln * AS + kk + bk]);
      v2f a1 = *(const v2f*)(&a_lds[(16 + ln) * AS + kk + bk]);
      acc0 = wmma4(a0, b01, acc0);
      acc1 = wmma4(a1, b01, acc1);
      a0 = *(const v2f*)(&a_lds[ln * AS + kk + 4 + bk]);
      a1 = *(const v2f*)(&a_lds[(16 + ln) * AS + kk + 4 + bk]);
      acc0 = wmma4(a0, b23, acc0);
      acc1 = wmma4(a1, b23, acc1);
    }
  }

  float bb = bias[n0 + ln];
  float s = 0.f, q = 0.f;
  #pragma unroll
  for (int v = 0; v < 8; ++v) {
    float val = acc0[v] + bb;
    if (postRelu) val = fmaxf(val, 0.f);
    int gr = r0 + v + 8 * hi;
    if (gr < M) {
      Y[(size_t)gr * FDIM + n0 + ln] = val;
      s += val; q += val * val;
    }
    val = acc1[v] + bb;
    if (postRelu) val = fmaxf(val, 0.f);
    gr = r0 + 16 + v + 8 * hi;
    if (gr < M) {
      Y[(size_t)gr * FDIM + n0 + ln] = val;
      s += val; q += val * val;
    }
  }
  if (psum) {
    s += __shfl_xor(s, 16, 32);
    q += __shfl_xor(q, 16, 32);
    if (hi == 0) {
      psum[(size_t)blockIdx.x * FDIM + n0 + ln] = s;
      psq [(size_t)blockIdx.x * FDIM + n0 + ln] = q;
    }
  }
}

// ---------------------------------------------------------------------------
// Reduce BN partials -> per-column fused scale/shift (deterministic tree)
// grid = 128 blocks (one per column)
// ---------------------------------------------------------------------------
__global__ __launch_bounds__(256)
void bn_stats_kernel(const float* __restrict__ psum, const float* __restrict__ psq,
                     int nblk, int M,
                     const float* __restrict__ gamma, const float* __restrict__ beta,
                     float* __restrict__ aOut, float* __restrict__ bOut)
{
  __shared__ float ss[256], sq[256];
  const int c = blockIdx.x;
  float s = 0.f, q = 0.f;
  for (int b = threadIdx.x; b < nblk; b += 256) {
    s += psum[(size_t)b * FDIM + c];
    q += psq [(size_t)b * FDIM + c];
  }
  ss[threadIdx.x] = s; sq[threadIdx.x] = q;
  __syncthreads();
  for (int off = 128; off > 0; off >>= 1) {
    if (threadIdx.x < off) {
      ss[threadIdx.x] += ss[threadIdx.x + off];
      sq[threadIdx.x] += sq[threadIdx.x + off];
    }
    __syncthreads();
  }
  if (threadIdx.x == 0) {
    float inv  = 1.f / (float)M;
    float mean = ss[0] * inv;
    float var  = fmaxf(sq[0] * inv - mean * mean, 0.f);
    float a    = gamma[c] * rsqrtf(var + BN_EPS);
    aOut[c] = a;
    bOut[c] = beta[c] - mean * a;
  }
}

// ---------------------------------------------------------------------------
// Mean pool per graph (batch is sorted -> deterministic range sums)
// grid = G blocks, 128 threads (one per feature)
// ---------------------------------------------------------------------------
__global__ __launch_bounds__(128)
void pool_mean_kernel(const float* __restrict__ x, const int* __restrict__ batch,
                      float* __restrict__ pooled)
{
  const int g = blockIdx.x;
  __shared__ int s_lo, s_hi;
  if (threadIdx.x == 0) {
    int lo = 0, hi = N_NODES;
    while (lo < hi) { int mid = (lo + hi) >> 1; if (batch[mid] < g) lo = mid + 1; else hi = mid; }
    s_lo = lo;
    hi = N_NODES;
    while (lo < hi) { int mid = (lo + hi) >> 1; if (batch[mid] < g + 1) lo = mid + 1; else hi = mid; }
    s_hi = lo;
  }
  __syncthreads();
  const int lo = s_lo, hiN = s_hi;
  float s = 0.f;
  for (int n = lo; n < hiN; ++n) s += x[(size_t)n * FDIM + threadIdx.x];
  float cnt = (float)(hiN - lo);
  pooled[(size_t)g * FDIM + threadIdx.x] = s / fmaxf(cnt, 1.f);
}

__global__ __launch_bounds__(256)
void fill_zero_kernel(float* __restrict__ p, int n) {
  int i = blockIdx.x * 256 + threadIdx.x;
  if (i < n) p[i] = 0.f;
}

__global__ __launch_bounds__(256)
void copy_kernel(const float* __restrict__ a, float* __restrict__ b, int n) {
  int i = blockIdx.x * 256 + threadIdx.x;
  if (i < n) b[i] = a[i];
}

// ---------------------------------------------------------------------------
extern "C" void kernel_launch(void* const* d_in, const int* in_sizes, int n_in,
                              void* d_out, int out_size, void* d_ws, size_t ws_size,
                              hipStream_t stream)
{
  const float* x     = (const float*)d_in[0];
  const int*   ei    = (const int*)  d_in[1];   // [2,E] int32
  const float* eattr = (const float*)d_in[2];
  const int*   batch = (const int*)  d_in[3];
  const float* eW1   = (const float*)d_in[4];   // [L,228,128]
  const float* eb1   = (const float*)d_in[5];   // [L,128]
  const float* eW2   = (const float*)d_in[6];   // [L,128,128]
  const float* eb2   = (const float*)d_in[7];
  const float* nW1   = (const float*)d_in[8];   // [L,128,128]
  const float* nb1   = (const float*)d_in[9];
  const float* bn_g  = (const float*)d_in[10];
  const float* bn_b  = (const float*)d_in[11];
  const float* nW2   = (const float*)d_in[12];
  const float* nb2   = (const float*)d_in[13];
  const float* fW1   = (const float*)d_in[14];
  const float* fb1   = (const float*)d_in[15];
  const float* fg    = (const float*)d_in[16];
  const float* fbe   = (const float*)d_in[17];
  const float* fW2   = (const float*)d_in[18];
  const float* fb2   = (const float*)d_in[19];
  float* out = (float*)d_out;

  const int* src = ei;
  const int* dst = ei + N_EDGES;

  const size_t NF = (size_t)N_NODES * FDIM;
  const int nBlkN = (N_NODES + 31) / 32;        // 1563
  float* ws     = (float*)d_ws;
  float* xbuf   = ws;  ws += NF;
  float* aggr   = ws;  ws += NF;
  float* tbuf   = ws;  ws += NF;
  float* psum   = ws;  ws += (size_t)nBlkN * FDIM;
  float* psq    = ws;  ws += (size_t)nBlkN * FDIM;
  float* bnA    = ws;  ws += FDIM;
  float* bnB    = ws;  ws += FDIM;
  float* pooled = ws;  ws += (size_t)GNUM * FDIM;
  float* t2     = ws;  ws += (size_t)GNUM * FDIM;
  // repacked weights (fragment order)
  float* eW1p   = ws;  ws += (size_t)NLAYERS * K1P * FDIM;
  float* eW2p   = ws;  ws += (size_t)NLAYERS * FDIM * FDIM;
  float* nW1p   = ws;  ws += (size_t)NLAYERS * FDIM * FDIM;
  float* nW2p   = ws;  ws += (size_t)NLAYERS * FDIM * FDIM;
  float* fW1p   = ws;  ws += (size_t)FDIM * FDIM;
  float* fW2p   = ws;  ws += (size_t)FDIM * FDIM;

  // ---- repack all weight matrices into WMMA fragment order ----
  for (int l = 0; l < NLAYERS; ++l) {
    repack_w_kernel<<<(K1P * 32 + 255) / 256, 256, 0, stream>>>(
        eW1 + (size_t)l * K1 * FDIM, eW1p + (size_t)l * K1P * FDIM, K1, K1P);
    repack_w_kernel<<<(FDIM * 32 + 255) / 256, 256, 0, stream>>>(
        eW2 + (size_t)l * FDIM * FDIM, eW2p + (size_t)l * FDIM * FDIM, FDIM, FDIM);
    repack_w_kernel<<<(FDIM * 32 + 255) / 256, 256, 0, stream>>>(
        nW1 + (size_t)l * FDIM * FDIM, nW1p + (size_t)l * FDIM * FDIM, FDIM, FDIM);
    repack_w_kernel<<<(FDIM * 32 + 255) / 256, 256, 0, stream>>>(
        nW2 + (size_t)l * FDIM * FDIM, nW2p + (size_t)l * FDIM * FDIM, FDIM, FDIM);
  }
  repack_w_kernel<<<(FDIM * 32 + 255) / 256, 256, 0, stream>>>(fW1, fW1p, FDIM, FDIM);
  repack_w_kernel<<<(FDIM * 32 + 255) / 256, 256, 0, stream>>>(fW2, fW2p, FDIM, FDIM);

  const int nE = (int)NF;
  copy_kernel<<<(nE + 255) / 256, 256, 0, stream>>>(x, xbuf, nE);

  for (int l = 0; l < NLAYERS; ++l) {
    fill_zero_kernel<<<(nE + 255) / 256, 256, 0, stream>>>(aggr, nE);
    edge_mlp_kernel<<<N_EDGES / 32, 256, 0, stream>>>(
        xbuf, eattr, src, dst,
        eW1p + (size_t)l * K1P * FDIM, eb1 + (size_t)l * FDIM,
        eW2p + (size_t)l * FDIM * FDIM, eb2 + (size_t)l * FDIM, aggr);
    // t = (x + aggr) @ nW1 + nb1 ; collect BN partials
    gemm32_kernel<<<nBlkN, 256, 0, stream>>>(
        xbuf, aggr, nullptr, nullptr,
        nW1p + (size_t)l * FDIM * FDIM, nb1 + (size_t)l * FDIM,
        tbuf, psum, psq, 0, N_NODES);
    bn_stats_kernel<<<FDIM, 256, 0, stream>>>(
        psum, psq, nBlkN, N_NODES,
        bn_g + (size_t)l * FDIM, bn_b + (size_t)l * FDIM, bnA, bnB);
    // x = ReLU( ReLU(BN(t)) @ nW2 + nb2 )
    gemm32_kernel<<<nBlkN, 256, 0, stream>>>(
        tbuf, nullptr, bnA, bnB,
        nW2p + (size_t)l * FDIM * FDIM, nb2 + (size_t)l * FDIM,
        xbuf, nullptr, nullptr, 1, N_NODES);
  }

  pool_mean_kernel<<<GNUM, 128, 0, stream>>>(xbuf, batch, pooled);

  // t2 = pooled @ fW1 + fb1 ; BN partials over G rows
  gemm32_kernel<<<GNUM / 32, 256, 0, stream>>>(
      pooled, nullptr, nullptr, nullptr, fW1p, fb1, t2, psum, psq, 0, GNUM);
  bn_stats_kernel<<<FDIM, 256, 0, stream>>>(psum, psq, GNUM / 32, GNUM, fg, fbe, bnA, bnB);
  // out = ReLU(BN(t2)) @ fW2 + fb2   (no final ReLU)
  gemm32_kernel<<<GNUM / 32, 256, 0, stream>>>(
      t2, nullptr, bnA, bnB, fW2p, fb2, out, nullptr, nullptr, 0, GNUM);
}